// BotRGCN_80573586473708
// MI455X (gfx1250) — compile-verified
//
#include <hip/hip_runtime.h>
#include <hip/hip_bf16.h>

#define N_NODES 100000
#define EMB 192
#define THIRD 64
#define SLOPE 0.01f
#define KCHUNK 192   // K is always a multiple of 192 here (192 or 768)

typedef __attribute__((ext_vector_type(2))) float v2f;
typedef __attribute__((ext_vector_type(8))) float v8f;

__device__ __forceinline__ float lrelu(float v) { return v >= 0.0f ? v : SLOPE * v; }

// C[m0..m0+15, ccol0+n0..+15] (op)= act(A[16xK] @ B[Kx16] + bias)
// fp32 WMMA 16x16x4; one wave computes one 16x16 tile; 4 waves/block share
// the same B column tile, staged cooperatively in LDS per K-chunk.
// A-frag (ISA 7.12.2): lane l<16 holds A[m0+l][k+0..1]; lanes 16-31 hold K+2..3.
// B-frag: VGPR v, lanes 0-15 -> K=k+v, lanes 16-31 -> K=k+v+2, N=n0+(l&15).
// C/D: VGPR i -> row m0+i (lanes 0-15) / m0+i+8 (lanes 16-31), col n0+(l&15).
__global__ __launch_bounds__(128)
void wmma_gemm_f32(const float* __restrict__ A, int lda,
                   const float* __restrict__ B, int ldb,
                   const float* __restrict__ bias,
                   float* __restrict__ C, int ldc, int ccol0,
                   int M, int K, int accumulate, int act) {
  __shared__ float Bs[KCHUNK * 16];                    // 12 KB
  const int lane  = threadIdx.x;                       // 0..31
  const int wave  = threadIdx.y;                       // 0..3
  const int mtile = blockIdx.y * 4 + wave;
  const bool activeWave = (mtile * 16 < M);            // uniform per wave
  const int m0   = mtile * 16;
  const int n0   = blockIdx.x * 16;
  const int half = lane >> 4;                          // 0 | 1
  const int l    = lane & 15;
  const int tid  = wave * 32 + lane;

  v8f acc = {};
  const float* Arow = activeWave ? (A + (size_t)(m0 + l) * lda + 2 * half) : A;

  for (int k0 = 0; k0 < K; k0 += KCHUNK) {
    // stage B[k0:k0+KCHUNK, n0:n0+16] -> LDS (all 128 threads)
    #pragma unroll 4
    for (int idx = tid; idx < KCHUNK * 16; idx += 128) {
      int kk = idx >> 4, nn = idx & 15;
      Bs[idx] = B[(size_t)(k0 + kk) * ldb + n0 + nn];
    }
    __syncthreads();
    if (activeWave) {
      const float* Ak = Arow + k0;
      #pragma unroll 8
      for (int kk = 0; kk < KCHUNK; kk += 4) {
        v2f a, b;
        a.x = Ak[kk];                                  // merges to global_load_b64
        a.y = Ak[kk + 1];
        b.x = Bs[(kk + 2 * half) * 16 + l];            // ds_load, conflict-free
        b.y = Bs[(kk + 2 * half + 1) * 16 + l];
        acc = __builtin_amdgcn_wmma_f32_16x16x4_f32(
            /*neg_a=*/false, a, /*neg_b=*/false, b,
            /*c_mod=*/(short)0, acc, /*reuse_a=*/false, /*reuse_b=*/false);
      }
    }
    __syncthreads();
  }
  if (!activeWave) return;

  const float bv = bias ? bias[n0 + l] : 0.0f;
  float* Cp = C + (size_t)(m0 + 8 * half) * ldc + ccol0 + n0 + l;
  #pragma unroll
  for (int i = 0; i < 8; ++i) {
    float v = acc[i] + bv;
    if (act) v = lrelu(v);
    float* cp = Cp + (size_t)i * ldc;
    if (accumulate) *cp += v;
    else            *cp  = v;
  }
}

// num_prop (K=4) and cat_prop (K=3) projections -> X[:,64:128], X[:,128:192]
__global__ void proj_small(const float* __restrict__ num_prop,
                           const float* __restrict__ cat_prop,
                           const float* __restrict__ Wn, const float* __restrict__ bn,
                           const float* __restrict__ Wc, const float* __restrict__ bc,
                           float* __restrict__ X) {
  int t = blockIdx.x * blockDim.x + threadIdx.x;
  if (t >= N_NODES * THIRD) return;
  int i = t / THIRD, j = t % THIRD;
  const float* np = num_prop + (size_t)i * 4;
  float a = bn[j];
  #pragma unroll
  for (int k = 0; k < 4; ++k) a += np[k] * Wn[k * THIRD + j];
  X[(size_t)i * EMB + THIRD + j] = lrelu(a);
  const float* cp = cat_prop + (size_t)i * 3;
  float c = bc[j];
  #pragma unroll
  for (int k = 0; k < 3; ++k) c += cp[k] * Wc[k * THIRD + j];
  X[(size_t)i * EMB + 2 * THIRD + j] = lrelu(c);
}

// One wave per edge: scatter x[src] into per-relation accumulator at dst, count edges.
__global__ void edge_agg(const float* __restrict__ x,
                         const int* __restrict__ src, const int* __restrict__ dst,
                         const int* __restrict__ etype,
                         float* __restrict__ s0, float* __restrict__ s1,
                         float* __restrict__ c0, float* __restrict__ c1, int E) {
  int t = blockIdx.x * blockDim.x + threadIdx.x;
  int e = t >> 5;
  if (e >= E) return;
  int lane = t & 31;
  int si = src[e], di = dst[e], r = etype[e];
  const float* xr = x + (size_t)si * EMB;
  float* Sr = (r == 0 ? s0 : s1) + (size_t)di * EMB;
  #pragma unroll
  for (int j = 0; j < EMB / 32; ++j)
    atomicAdd(&Sr[lane + 32 * j], xr[lane + 32 * j]);
  if (lane == 0) atomicAdd((r == 0 ? c0 : c1) + di, 1.0f);
}

// agg_r = s_r / max(cnt_r, 1)
__global__ void finalize_agg(float* __restrict__ s0, float* __restrict__ s1,
                             const float* __restrict__ c0, const float* __restrict__ c1) {
  int t = blockIdx.x * blockDim.x + threadIdx.x;
  if (t >= N_NODES * EMB) return;
  int i = t / EMB;
  s0[t] /= fmaxf(c0[i], 1.0f);
  s1[t] /= fmaxf(c1[i], 1.0f);
}

// out[i, 0:2] = x[i,:] @ W_o2 + b_o2
__global__ void final_out(const float* __restrict__ X,
                          const float* __restrict__ W, const float* __restrict__ b,
                          float* __restrict__ out) {
  int i = blockIdx.x * blockDim.x + threadIdx.x;
  if (i >= N_NODES) return;
  const float* xr = X + (size_t)i * EMB;
  float a0 = b[0], a1 = b[1];
  for (int k = 0; k < EMB; ++k) {
    float v = xr[k];
    a0 += v * W[k * 2 + 0];
    a1 += v * W[k * 2 + 1];
  }
  out[(size_t)i * 2 + 0] = a0;
  out[(size_t)i * 2 + 1] = a1;
}

extern "C" void kernel_launch(void* const* d_in, const int* in_sizes, int n_in,
                              void* d_out, int out_size, void* d_ws, size_t ws_size,
                              hipStream_t stream) {
  const float* des       = (const float*)d_in[0];
  // d_in[1] = tweet: unused by the reference
  const float* num_prop  = (const float*)d_in[2];
  const float* cat_prop  = (const float*)d_in[3];
  const int*   edge_index= (const int*)d_in[4];
  const int*   edge_type = (const int*)d_in[5];
  const float* W_des     = (const float*)d_in[6];
  const float* b_des     = (const float*)d_in[7];
  const float* W_num     = (const float*)d_in[8];
  const float* b_num     = (const float*)d_in[9];
  const float* W_cat     = (const float*)d_in[10];
  const float* b_cat     = (const float*)d_in[11];
  const float* W_in      = (const float*)d_in[12];
  const float* b_in      = (const float*)d_in[13];
  const float* W_root    = (const float*)d_in[14];
  const float* W_rel     = (const float*)d_in[15];
  const float* b_rgcn    = (const float*)d_in[16];
  const float* W_o1      = (const float*)d_in[17];
  const float* b_o1      = (const float*)d_in[18];
  const float* W_o2      = (const float*)d_in[19];
  const float* b_o2      = (const float*)d_in[20];
  float* out = (float*)d_out;

  const int E = in_sizes[5];
  const int* src = edge_index;
  const int* dst = edge_index + E;

  char* ws = (char*)d_ws;
  const size_t bufBytes = (size_t)N_NODES * EMB * sizeof(float);
  float* X0 = (float*)(ws);
  float* X1 = (float*)(ws + bufBytes);
  float* S0 = (float*)(ws + 2 * bufBytes);
  float* S1 = (float*)(ws + 3 * bufBytes);
  float* C0 = (float*)(ws + 4 * bufBytes);
  float* C1 = C0 + N_NODES;

  auto gemm = [&](const float* A, int lda, const float* B, int ldb, const float* bias,
                  float* C, int ldc, int ccol0, int M, int Ncols, int K,
                  int accum, int act) {
    dim3 block(32, 4);
    dim3 grid(Ncols / 16, (M / 16 + 3) / 4);
    wmma_gemm_f32<<<grid, block, 0, stream>>>(A, lda, B, ldb, bias, C, ldc, ccol0,
                                              M, K, accum, act);
  };

  auto rgcn = [&](const float* x, float* o) {
    hipMemsetAsync(S0, 0, bufBytes, stream);
    hipMemsetAsync(S1, 0, bufBytes, stream);
    hipMemsetAsync(C0, 0, 2 * (size_t)N_NODES * sizeof(float), stream);
    // o = x @ W_root + b_rgcn
    gemm(x, EMB, W_root, EMB, b_rgcn, o, EMB, 0, N_NODES, EMB, EMB, 0, 0);
    // per-relation mean aggregation
    int tot = E * 32;
    edge_agg<<<(tot + 255) / 256, 256, 0, stream>>>(x, src, dst, edge_type,
                                                    S0, S1, C0, C1, E);
    int tot2 = N_NODES * EMB;
    finalize_agg<<<(tot2 + 255) / 256, 256, 0, stream>>>(S0, S1, C0, C1);
    // o += agg_r @ W_rel[r]
    gemm(S0, EMB, W_rel,             EMB, nullptr, o, EMB, 0, N_NODES, EMB, EMB, 1, 0);
    gemm(S1, EMB, W_rel + EMB * EMB, EMB, nullptr, o, EMB, 0, N_NODES, EMB, EMB, 1, 0);
  };

  // 1) feature projections -> X0 [N,192]
  gemm(des, 768, W_des, THIRD, b_des, X0, EMB, 0, N_NODES, THIRD, 768, 0, 1);
  proj_small<<<(N_NODES * THIRD + 255) / 256, 256, 0, stream>>>(
      num_prop, cat_prop, W_num, b_num, W_cat, b_cat, X0);

  // 2) X1 = LReLU(X0 @ W_in + b_in)
  gemm(X0, EMB, W_in, EMB, b_in, X1, EMB, 0, N_NODES, EMB, EMB, 0, 1);

  // 3) two RGCN convolutions
  rgcn(X1, X0);
  rgcn(X0, X1);

  // 4) X0 = LReLU(X1 @ W_o1 + b_o1)
  gemm(X1, EMB, W_o1, EMB, b_o1, X0, EMB, 0, N_NODES, EMB, EMB, 0, 1);

  // 5) out = X0 @ W_o2 + b_o2
  final_out<<<(N_NODES + 255) / 256, 256, 0, stream>>>(X0, W_o2, b_o2, out);
}